// ProposalLayer_3977139716956
// MI455X (gfx1250) — compile-verified
//
#include <hip/hip_runtime.h>
#include <stdint.h>

#define B_IMG 8
#define N_ANC 250000
#define PRE_NMS 2000
#define POST_NMS 1000
#define CAP 4096
#define MW 63                 // ceil(2000/32) mask words per row
#define HBINS 65536           // 16-bit histogram bins
#define NMS_THRESH 0.7f
#define MIN_BOX 4.0f
#define XFORM_CLIP 4.135f

// ---------- helpers ----------
__device__ __forceinline__ unsigned float_to_ordered(float f) {
    unsigned b = __float_as_uint(f);
    return (b & 0x80000000u) ? ~b : (b | 0x80000000u);
}
// u(-inf) = ~0xFF800000 = 0x007FFFFF ; valid <=> u > that
#define U_NEGINF 0x007FFFFFu

__device__ __forceinline__ float4 decode_box(float4 a, float4 d, float Wf, float Hf) {
    float aw = a.z - a.x, ah = a.w - a.y;
    float ax = a.x + 0.5f * aw, ay = a.y + 0.5f * ah;
    float dw = fminf(d.z, XFORM_CLIP), dh = fminf(d.w, XFORM_CLIP);
    float px = d.x * aw + ax, py = d.y * ah + ay;
    float pw = expf(dw) * aw, ph = expf(dh) * ah;
    float4 o;
    o.x = fminf(fmaxf(px - 0.5f * pw, 0.f), Wf);
    o.y = fminf(fmaxf(py - 0.5f * ph, 0.f), Hf);
    o.z = fminf(fmaxf(px + 0.5f * pw, 0.f), Wf);
    o.w = fminf(fmaxf(py + 0.5f * ph, 0.f), Hf);
    return o;
}

// ---------- 1: decode + size filter -> ordered-uint scores + 16-bit histogram ----------
__global__ void k_score_hist(const float4* __restrict__ pred, const float* __restrict__ obj,
                             const float4* __restrict__ anch, const int* __restrict__ hp,
                             const int* __restrict__ wp, unsigned* __restrict__ uscore,
                             unsigned* __restrict__ hist) {
    int i = blockIdx.x * blockDim.x + threadIdx.x;
    if (i >= B_IMG * N_ANC) return;
    int b = i / N_ANC;
    int n = i - b * N_ANC;
    float Hf = (float)hp[0], Wf = (float)wp[0];
    float4 box = decode_box(anch[n], pred[(size_t)b * N_ANC + n], Wf, Hf);
    bool ok = ((box.z - box.x) >= MIN_BOX) && ((box.w - box.y) >= MIN_BOX);
    float s = ok ? obj[i] : -__builtin_inff();
    unsigned u = float_to_ordered(s);
    uscore[i] = u;
    atomicAdd(&hist[(size_t)b * HBINS + (u >> 16)], 1u);
}

// ---------- 2: per-image suffix scan of 65536 bins -> threshold (bin<<16) ----------
__global__ void k_scan(const unsigned* __restrict__ hist, unsigned* __restrict__ thr) {
    __shared__ unsigned chunk[1024];
    int b = blockIdx.x, tid = threadIdx.x;
    const unsigned* h = hist + (size_t)b * HBINS;
    unsigned s = 0;
    #pragma unroll 4
    for (int j = 0; j < 64; ++j) s += h[tid * 64 + j];
    chunk[tid] = s;
    __syncthreads();
    if (tid == 0) {
        const unsigned K = PRE_NMS;
        unsigned acc = 0;
        int cidx = 0;
        for (int t = 1023; t >= 0; --t) {
            if (acc + chunk[t] >= K) { cidx = t; break; }
            acc += chunk[t];
        }
        unsigned bin = 0;
        for (int j = 63; j >= 0; --j) {
            unsigned c = h[cidx * 64 + j];
            if (acc + c >= K) { bin = (unsigned)(cidx * 64 + j); break; }
            acc += c;
        }
        thr[b] = bin << 16;   // count(u >= thr) in [2000, 2000 + hist[bin])
    }
}

// ---------- 3: compact candidates (u >= T) as 64-bit sort keys ----------
__global__ void k_compact(const unsigned* __restrict__ uscore, const unsigned* __restrict__ thr,
                          unsigned* __restrict__ cnt, unsigned long long* __restrict__ keys) {
    int i = blockIdx.x * blockDim.x + threadIdx.x;
    if (i >= B_IMG * N_ANC) return;
    int b = i / N_ANC;
    int n = i - b * N_ANC;
    unsigned u = uscore[i];
    if (u >= thr[b]) {
        unsigned pos = atomicAdd(&cnt[b], 1u);
        if (pos < CAP)
            keys[(size_t)b * CAP + pos] = (((unsigned long long)(~u)) << 32) | (unsigned)n;
    }
}

// ---------- 4: bitonic sort 4096 keys in LDS, decode top-2000 ----------
__global__ void k_sort_decode(const unsigned long long* __restrict__ keys,
                              const unsigned* __restrict__ cnt,
                              const float4* __restrict__ pred, const float4* __restrict__ anch,
                              const int* __restrict__ hp, const int* __restrict__ wp,
                              float4* __restrict__ tboxes, unsigned* __restrict__ tvalid) {
    __shared__ unsigned long long sk[CAP];  // 32 KB
    int b = blockIdx.x, tid = threadIdx.x, bd = blockDim.x;
    unsigned m = cnt[b]; if (m > CAP) m = CAP;
    for (int j = tid; j < CAP; j += bd)
        sk[j] = (j < (int)m) ? keys[(size_t)b * CAP + j] : ~0ULL;
    __syncthreads();
    for (unsigned kk = 2; kk <= CAP; kk <<= 1) {
        for (unsigned jj = kk >> 1; jj > 0; jj >>= 1) {
            for (int i = tid; i < CAP; i += bd) {
                int ixj = i ^ (int)jj;
                if (ixj > i) {
                    bool up = ((i & (int)kk) == 0);
                    unsigned long long a = sk[i], c = sk[ixj];
                    if (up ? (a > c) : (a < c)) { sk[i] = c; sk[ixj] = a; }
                }
            }
            __syncthreads();
        }
    }
    float Hf = (float)hp[0], Wf = (float)wp[0];
    for (int j = tid; j < PRE_NMS; j += bd) {
        unsigned long long key = sk[j];
        float4 box = make_float4(0.f, 0.f, 0.f, 0.f);
        unsigned v = 0u;
        if (key != ~0ULL) {
            unsigned u = ~(unsigned)(key >> 32);
            unsigned n = (unsigned)key;
            v = (u > U_NEGINF) ? 1u : 0u;
            box = decode_box(anch[n], pred[(size_t)b * N_ANC + n], Wf, Hf);
        }
        tboxes[(size_t)b * PRE_NMS + j] = box;
        tvalid[(size_t)b * PRE_NMS + j] = v;
    }
}

// ---------- 5: suppression masks; 8 row-tiles per block share one async-staged col tile ----------
__global__ void __launch_bounds__(256) k_iou(const float4* __restrict__ boxes,
                                             unsigned* __restrict__ mask) {
    __shared__ float4 cb[32];
    int tid = threadIdx.x;
    int lane = tid & 31;
    int wave = tid >> 5;
    int b = blockIdx.z;
    int cbase = blockIdx.x * 32;
    if (wave == 0) {
        // whole wave active -> EXEC all-ones for the async op; clamp index instead of masking
        int c = min(cbase + lane, PRE_NMS - 1);
        const float4* gp = &boxes[(size_t)b * PRE_NMS + c];
        unsigned lds_off = (unsigned)(unsigned long long)(uintptr_t)(&cb[lane]);
        // CDNA5 async global->LDS copy (ASYNCcnt), one 16B column box per lane
        asm volatile("global_load_async_to_lds_b128 %0, %1, off"
                     :: "v"(lds_off), "v"(gp) : "memory");
        asm volatile("s_wait_asynccnt 0" ::: "memory");
    }
    __syncthreads();
    int row = (blockIdx.y * 8 + wave) * 32 + lane;
    if (row < PRE_NMS) {
        float4 rb = boxes[(size_t)b * PRE_NMS + row];
        float ra = (rb.z - rb.x) * (rb.w - rb.y);
        unsigned bits = 0u;
        #pragma unroll
        for (int k = 0; k < 32; ++k) {
            float4 c4 = cb[k];
            int cidx = cbase + k;
            float ca = (c4.z - c4.x) * (c4.w - c4.y);
            float iw = fmaxf(fminf(rb.z, c4.z) - fmaxf(rb.x, c4.x), 0.f);
            float ih = fmaxf(fminf(rb.w, c4.w) - fmaxf(rb.y, c4.y), 0.f);
            float inter = iw * ih;
            float iou = inter / (ra + ca - inter + 1e-9f);
            if ((iou > NMS_THRESH) && (cidx > row) && (cidx < PRE_NMS)) bits |= (1u << k);
        }
        mask[((size_t)b * PRE_NMS + row) * MW + blockIdx.x] = bits;
    }
}

// ---------- 6: serial greedy collect of kept indices ----------
__global__ void k_collect(const unsigned* __restrict__ mask, const unsigned* __restrict__ tvalid,
                          int* __restrict__ sel, int* __restrict__ nkeep) {
    __shared__ unsigned remv[MW];
    __shared__ int cnt_s;
    __shared__ int curKeep;
    int b = blockIdx.x, tid = threadIdx.x, bd = blockDim.x;
    for (int w = tid; w < MW; w += bd) {
        unsigned m = 0u;
        for (int l = 0; l < 32; ++l) {
            int e = w * 32 + l;
            if (e >= PRE_NMS || tvalid[(size_t)b * PRE_NMS + e] == 0u) m |= (1u << l);
        }
        remv[w] = m;
    }
    if (tid == 0) cnt_s = 0;
    __syncthreads();
    for (int i = 0; i < PRE_NMS; ++i) {
        if (tid == 0) {
            int kept = !((remv[i >> 5] >> (i & 31)) & 1u);
            curKeep = kept;
            if (kept) {
                if (cnt_s < POST_NMS) sel[b * POST_NMS + cnt_s] = i;
                cnt_s++;
            }
        }
        __syncthreads();
        if (curKeep) {
            for (int w = tid; w < MW; w += bd)
                remv[w] |= mask[((size_t)b * PRE_NMS + i) * MW + w];
        }
        __syncthreads();
    }
    if (tid == 0) nkeep[b] = (cnt_s < POST_NMS) ? cnt_s : POST_NMS;
}

// ---------- 7: emit boxes + valid flags ----------
__global__ void k_output(const float4* __restrict__ tboxes, const int* __restrict__ sel,
                         const int* __restrict__ nkeep, float* __restrict__ out) {
    int j = blockIdx.x * blockDim.x + threadIdx.x;
    if (j >= B_IMG * POST_NMS) return;
    int b = j / POST_NMS;
    int p = j - b * POST_NMS;
    float4* outb = (float4*)out;                 // B*POST float4
    float* outv = out + (size_t)B_IMG * POST_NMS * 4;
    if (p < nkeep[b]) {
        outb[j] = tboxes[(size_t)b * PRE_NMS + sel[b * POST_NMS + p]];
        outv[j] = 1.0f;
    } else {
        outb[j] = make_float4(0.f, 0.f, 0.f, 0.f);
        outv[j] = 0.0f;
    }
}

static inline size_t align256(size_t x) { return (x + 255) & ~(size_t)255; }

extern "C" void kernel_launch(void* const* d_in, const int* in_sizes, int n_in,
                              void* d_out, int out_size, void* d_ws, size_t ws_size,
                              hipStream_t stream) {
    const float4* pred = (const float4*)d_in[0];   // (8,250000,4) f32
    const float*  obj  = (const float*)d_in[1];    // (8,250000)  f32
    const float4* anch = (const float4*)d_in[2];   // (250000,4)  f32
    const int*    hp   = (const int*)d_in[3];      // scalar 800
    const int*    wp   = (const int*)d_in[4];      // scalar 1333
    (void)in_sizes; (void)n_in; (void)out_size; (void)ws_size;

    // workspace layout (~15 MB)
    char* w = (char*)d_ws;
    size_t off = 0;
    unsigned* uscore = (unsigned*)(w + off);             off = align256(off + (size_t)B_IMG * N_ANC * 4);
    unsigned* hist   = (unsigned*)(w + off);             off = align256(off + (size_t)B_IMG * HBINS * 4);
    unsigned* thr    = (unsigned*)(w + off);             off = align256(off + (size_t)B_IMG * 4);
    unsigned* cnt    = (unsigned*)(w + off);             off = align256(off + (size_t)B_IMG * 4);
    unsigned long long* keys = (unsigned long long*)(w + off); off = align256(off + (size_t)B_IMG * CAP * 8);
    float4*   tboxes = (float4*)(w + off);               off = align256(off + (size_t)B_IMG * PRE_NMS * 16);
    unsigned* tvalid = (unsigned*)(w + off);             off = align256(off + (size_t)B_IMG * PRE_NMS * 4);
    unsigned* masks  = (unsigned*)(w + off);             off = align256(off + (size_t)B_IMG * PRE_NMS * MW * 4);
    int*      sel    = (int*)(w + off);                  off = align256(off + (size_t)B_IMG * POST_NMS * 4);
    int*      nkeep  = (int*)(w + off);                  off = align256(off + (size_t)B_IMG * 4);

    hipMemsetAsync(hist, 0, (size_t)B_IMG * HBINS * 4, stream);
    hipMemsetAsync(cnt, 0, (size_t)B_IMG * 4, stream);

    int total = B_IMG * N_ANC;
    k_score_hist<<<(total + 255) / 256, 256, 0, stream>>>(pred, obj, anch, hp, wp, uscore, hist);
    k_scan<<<B_IMG, 1024, 0, stream>>>(hist, thr);
    k_compact<<<(total + 255) / 256, 256, 0, stream>>>(uscore, thr, cnt, keys);
    k_sort_decode<<<B_IMG, 1024, 0, stream>>>(keys, cnt, pred, anch, hp, wp, tboxes, tvalid);
    k_iou<<<dim3(MW, (PRE_NMS + 255) / 256, B_IMG), 256, 0, stream>>>(tboxes, masks);
    k_collect<<<B_IMG, 64, 0, stream>>>(masks, tvalid, sel, nkeep);
    k_output<<<(B_IMG * POST_NMS + 255) / 256, 256, 0, stream>>>(tboxes, sel, nkeep, (float*)d_out);
}